// BinaryEncoding_16819091931479
// MI455X (gfx1250) — compile-verified
//
#include <hip/hip_runtime.h>

typedef unsigned int       u32;
typedef unsigned long long u64;
typedef __attribute__((ext_vector_type(4))) u32 v4u;
typedef __attribute__((ext_vector_type(4))) int v4i;
typedef __attribute__((ext_vector_type(8))) int v8i;

// Explicit LDS address space so tile reads lower to ds_load_b32 (not flat).
typedef __attribute__((address_space(3))) const float lds_cfloat;

#define N_B    32
#define N_C    256
#define HW     3136   // 56*56
#define TPIX   64     // pixels per block (3136 % 64 == 0 -> no tail)
#define HALF_C 128

// Issue one 2-D TDM tile load: tile_d0 contiguous elements (x) by tile_d1 rows (y),
// row stride = stride0 elements, 4-byte elements, into LDS at lds_byte_off.
// Descriptor packing per CDNA5 ISA 8.3 (group0) / 8.4 (group1).
__device__ __forceinline__ void tdm_load_tile_2d(u32 lds_byte_off, u64 gaddr_bytes,
                                                 u32 tensor_d0, u32 tensor_d1,
                                                 u32 tile_d0, u32 tile_d1,
                                                 u64 stride0_elems)
{
    v4u g0;
    g0.x = 1u;                                            // count=1, user descriptor
    g0.y = lds_byte_off;                                  // lds_addr [63:32]
    g0.z = (u32)(gaddr_bytes & 0xFFFFFFFFull);            // global_addr[31:0]
    g0.w = (u32)((gaddr_bytes >> 32) & 0x01FFFFFFull)     // global_addr[56:32]
         | (2u << 30);                                    // type=2 ("image")

    v8i g1;
    g1[0] = (int)(2u << 16);                                          // data_size=4B; mask/flags=0
    g1[1] = (int)((tensor_d0 & 0xFFFFu) << 16);                       // td0[15:0] @63:48
    g1[2] = (int)((tensor_d0 >> 16) | ((tensor_d1 & 0xFFFFu) << 16)); // td0[31:16], td1[15:0]
    g1[3] = (int)((tensor_d1 >> 16) | ((tile_d0 & 0xFFFFu) << 16));   // td1[31:16], tile_dim0
    g1[4] = (int)(tile_d1 & 0xFFFFu);                                 // tile_dim1; tile_dim2=0
    g1[5] = (int)(u32)(stride0_elems & 0xFFFFFFFFull);                // stride0[31:0]
    g1[6] = (int)(u32)((stride0_elems >> 32) & 0xFFFFull);            // stride0[47:32]; stride1=0
    g1[7] = 0;

    v4i gz = {0, 0, 0, 0};
#if defined(__clang_major__) && (__clang_major__ >= 23)
    v8i gz8 = {0, 0, 0, 0, 0, 0, 0, 0};
    __builtin_amdgcn_tensor_load_to_lds(g0, g1, gz, gz, gz8, 0);
#else
    __builtin_amdgcn_tensor_load_to_lds(g0, g1, gz, gz, 0);
#endif
}

__global__ __launch_bounds__(TPIX)
void BinaryEncoding_top16_kernel(const float* __restrict__ in, float* __restrict__ out)
{
    __shared__ float tile[N_C * TPIX];   // 64 KB: tile[c*64 + p]

    const int p   = (int)threadIdx.x;    // pixel within block, 0..63
    const u32 hw0 = (u32)blockIdx.x * TPIX;
    const int b   = (int)blockIdx.y;

    const u64 base_elem = ((u64)b * N_C) * (u64)HW + hw0;             // (b, c=0, hw0)
    const u64 gaddr     = (u64)(uintptr_t)in + base_elem * 4ull;
    const u32 lds0      = (u32)(uintptr_t)&tile[0];                   // low 32 bits = LDS offset

    if (threadIdx.x < 32) {  // wave 0 issues both DMA descriptors
        tdm_load_tile_2d(lds0, gaddr,
                         (u32)(HW - hw0), HALF_C, TPIX, HALF_C, HW);
        tdm_load_tile_2d(lds0 + (u32)(TPIX * HALF_C * 4),
                         gaddr + (u64)HALF_C * (u64)HW * 4ull,
                         (u32)(HW - hw0), HALF_C, TPIX, HALF_C, HW);
    }

    // Launder the LDS byte offset (scalar) so the compiler cannot reason about
    // the tile object (TDM writes it behind the compiler's back), then rebuild
    // an addrspace(3) pointer -> reads stay on the DS pipe (ds_load_b32).
    u32 lds_off = lds0;
    asm volatile("" : "+s"(lds_off));
    const lds_cfloat* t = (const lds_cfloat*)lds_off;

    // Streaming top-16 via med3 insertion chain (16 ops/element).
    float L[16];
#pragma unroll
    for (int k = 0; k < 16; ++k) L[k] = -__builtin_inff();

    __builtin_amdgcn_s_wait_tensorcnt(1);   // first half landed (no-op for waves w/o TDM)
    __syncthreads();

#pragma unroll 4
    for (int c = 0; c < HALF_C; ++c) {
        float v = t[c * TPIX + p];
#pragma unroll
        for (int k = 15; k >= 1; --k) L[k] = __builtin_amdgcn_fmed3f(L[k - 1], L[k], v);
        L[0] = fmaxf(L[0], v);
    }

    __builtin_amdgcn_s_wait_tensorcnt(0);   // second half landed
    __syncthreads();

#pragma unroll 4
    for (int c = HALF_C; c < N_C; ++c) {
        float v = t[c * TPIX + p];
#pragma unroll
        for (int k = 15; k >= 1; --k) L[k] = __builtin_amdgcn_fmed3f(L[k - 1], L[k], v);
        L[0] = fmaxf(L[0], v);
    }

    const float T = L[15];                  // 16th largest
    float* op = out + base_elem + (u64)p;
#pragma unroll 4
    for (int c = 0; c < N_C; ++c) {
        float v = t[c * TPIX + p];
        __builtin_nontemporal_store((v >= T) ? 1.0f : 0.0f, op + (u64)c * HW);
    }
}

extern "C" void kernel_launch(void* const* d_in, const int* in_sizes, int n_in,
                              void* d_out, int out_size, void* d_ws, size_t ws_size,
                              hipStream_t stream)
{
    const float* in  = (const float*)d_in[0];
    float*       out = (float*)d_out;
    dim3 grid(HW / TPIX, N_B);   // (49, 32)
    dim3 block(TPIX);            // 64 threads = 2 waves
    hipLaunchKernelGGL(BinaryEncoding_top16_kernel, grid, block, 0, stream, in, out);
}